// GaussianVectorQuantizer_90804198572213
// MI455X (gfx1250) — compile-verified
//
#include <hip/hip_runtime.h>

typedef __attribute__((ext_vector_type(2))) float v2f;
typedef __attribute__((ext_vector_type(8))) float v8f;

#define B_    8
#define C_    256
#define T_    4096
#define K_    1024
#define N_    (B_ * T_)
#define ROWS  32
#define ZLD   258      // padded stride for z / z_q tile (bank-conflict avoidance)
#define LLD   1028     // padded stride for logits tile
#define TEMP_INV 2.0f  // 1 / TEMPERATURE(0.5)
#define EPSQ  1e-10f

// d_out layout (floats): z_q | precision_q | prob | log_prob | mean_prob
#define ZQ_OFF    ((size_t)0)
#define PQ_OFF    ((size_t)8388608)
#define PROB_OFF  ((size_t)8388609)
#define LPROB_OFF ((size_t)41943041)
#define MPROB_OFF ((size_t)75497473)

__device__ __forceinline__ float wave_max(float v) {
  #pragma unroll
  for (int m = 16; m >= 1; m >>= 1) v = fmaxf(v, __shfl_xor(v, m, 32));
  return v;
}
__device__ __forceinline__ float wave_sum(float v) {
  #pragma unroll
  for (int m = 16; m >= 1; m >>= 1) v += __shfl_xor(v, m, 32);
  return v;
}

// ---- kernel 0: book row norms, zero mean accumulator, precision_q ----
__global__ void gvq_prep(const float* __restrict__ book,
                         const float* __restrict__ lpq,
                         float* __restrict__ ws_bsq,
                         float* __restrict__ ws_macc,
                         float* __restrict__ out_pq) {
  __shared__ float red[256];
  const int k = blockIdx.x, tid = threadIdx.x;
  float v = book[(size_t)k * C_ + tid];
  red[tid] = v * v;
  __syncthreads();
  for (int s = 128; s > 0; s >>= 1) {
    if (tid < s) red[tid] += red[tid + s];
    __syncthreads();
  }
  if (tid == 0) { ws_bsq[k] = red[0]; ws_macc[k] = 0.0f; }
  if (k == 0 && tid == 0) *out_pq = 0.5f / (1.0f + __expf(lpq[0]));
}

// ---- kernel 1: fused distance-GEMM + softmaxes + gumbel + zq-GEMM ----
__global__ void __launch_bounds__(256, 1)
gvq_main(const float* __restrict__ z,
         const float* __restrict__ book,
         const float* __restrict__ lpq,
         const float* __restrict__ u,
         const float* __restrict__ ws_bsq,
         float* __restrict__ ws_macc,
         float* __restrict__ out) {
  extern __shared__ float smem[];
  float* zt    = smem;                 // ROWS*ZLD  (reused as z_q staging later)
  float* znorm = zt + ROWS * ZLD;      // ROWS
  float* lg    = znorm + ROWS;         // ROWS*LLD  (logits -> encodings in place)

  const int tid  = threadIdx.x;
  const int lane = tid & 31;
  const int w    = tid >> 5;            // wave id 0..7
  const int n0   = blockIdx.x * ROWS;   // T_ % ROWS == 0, so b constant per tile
  const int b    = n0 / T_;
  const int t0   = n0 % T_;
  const float pq = 0.5f / (1.0f + __expf(lpq[0]));

  if (tid < ROWS) znorm[tid] = 0.0f;
  __syncthreads();

  // Load transposed z tile: zt[r][c] = z[(b*C + c)*T + t0 + r]  (coalesced over r)
  const int r = tid & 31;
  float sq = 0.0f;
  #pragma unroll
  for (int it = 0; it < 32; ++it) {
    const int c = it * 8 + (tid >> 5);
    const float v = z[((size_t)(b * C_ + c)) * T_ + t0 + r];
    zt[r * ZLD + c] = v;
    sq += v * v;
  }
  atomicAdd(&znorm[r], sq);
  __syncthreads();

  const v8f zero8 = {0.f, 0.f, 0.f, 0.f, 0.f, 0.f, 0.f, 0.f};
  const int m    = lane & 15;
  const int koff = (lane >> 4) * 2;

  // ---------- GEMM1: logits = -pq * (||z||^2 + ||e||^2 - 2 z.e) ----------
  // Each wave: 16 rows x 256 cols, done as TWO passes of 8 column-tiles so
  // only 64 accumulator VGPRs are live at a time (no scratch spilling).
  {
    const int rt = w >> 2;   // row-tile 0..1 (rows rt*16..rt*16+15)
    const int cg = w & 3;    // col-group: tiles cg*16 .. cg*16+15
    const float* zrow = &zt[(rt * 16 + m) * ZLD];

    #pragma unroll 1
    for (int pass = 0; pass < 2; ++pass) {
      v8f acc[8];
      #pragma unroll
      for (int j = 0; j < 8; ++j) acc[j] = zero8;

      const int tbase = cg * 16 + pass * 8;   // first column-tile of this pass
      #pragma unroll 1
      for (int k = 0; k < C_; k += 4) {
        v2f a; a.x = zrow[k + koff]; a.y = zrow[k + koff + 1];
        #pragma unroll
        for (int j = 0; j < 8; ++j) {
          const int kb = (tbase + j) * 16 + m;             // codebook entry (B col)
          const float* bp = &book[(size_t)kb * C_ + k + koff];
          v2f bb; bb.x = bp[0]; bb.y = bp[1];
          acc[j] = __builtin_amdgcn_wmma_f32_16x16x4_f32(
              false, a, false, bb, (short)0, acc[j], false, false);
        }
      }

      float zn[8];
      #pragma unroll
      for (int v = 0; v < 8; ++v) zn[v] = znorm[rt * 16 + v + ((lane >> 4) << 3)];

      #pragma unroll
      for (int j = 0; j < 8; ++j) {
        const int col = (tbase + j) * 16 + m;
        const float bsq = ws_bsq[col];
        #pragma unroll
        for (int v = 0; v < 8; ++v) {
          const int row = rt * 16 + v + ((lane >> 4) << 3);
          lg[row * LLD + col] = -pq * (zn[v] + bsq - 2.0f * acc[j][v]);
        }
      }
    }
  }
  __syncthreads();

  // ---------- softmax / log_softmax / gumbel-softmax (4 rows per wave) ----------
  float pacc[32];
  #pragma unroll
  for (int i = 0; i < 32; ++i) pacc[i] = 0.0f;

  #pragma unroll 1
  for (int i = 0; i < 4; ++i) {
    const int row = w * 4 + i;
    const size_t n = (size_t)(n0 + row);
    float* lrow = &lg[row * LLD];

    float mx = -3.4e38f;
    for (int c = lane; c < K_; c += 32) mx = fmaxf(mx, lrow[c]);
    mx = wave_max(mx);

    float s = 0.0f;
    for (int c = lane; c < K_; c += 32) s += __expf(lrow[c] - mx);
    s = wave_sum(s);
    const float ls = __logf(s);

    float m2 = -3.4e38f;
    #pragma unroll
    for (int ii = 0; ii < 32; ++ii) {
      const int c = lane + (ii << 5);
      const float l  = lrow[c];
      const float lp = l - mx - ls;
      const float p  = __expf(lp);
      out[PROB_OFF  + n * K_ + c] = p;
      out[LPROB_OFF + n * K_ + c] = lp;
      pacc[ii] += p;
      const float uu = u[n * K_ + c];
      const float g  = -__logf(-__logf(uu + EPSQ) + EPSQ);
      const float val = (l + g) * TEMP_INV;
      lrow[c] = val;
      m2 = fmaxf(m2, val);
    }
    m2 = wave_max(m2);

    float s2 = 0.0f;
    for (int c = lane; c < K_; c += 32) s2 += __expf(lrow[c] - m2);
    s2 = wave_sum(s2);
    const float ls2 = __logf(s2);
    for (int c = lane; c < K_; c += 32) lrow[c] = __expf(lrow[c] - m2 - ls2);
  }
  // fold this block's prob column-sums into global mean accumulator
  #pragma unroll
  for (int i = 0; i < 32; ++i) atomicAdd(&ws_macc[lane + (i << 5)], pacc[i]);
  __syncthreads();

  // ---------- GEMM2: z_q = encodings @ book ----------
  {
    const int rt2 = w >> 2;  // row-tile 0..1
    const int cq  = w & 3;   // owns channel tiles cq*4 .. cq*4+3
    v8f acc2[4];
    #pragma unroll
    for (int q = 0; q < 4; ++q) acc2[q] = zero8;

    const float* arow = &lg[(rt2 * 16 + m) * LLD];
    #pragma unroll 1
    for (int kb = 0; kb < K_; kb += 4) {
      v2f a; a.x = arow[kb + koff]; a.y = arow[kb + koff + 1];
      #pragma unroll
      for (int q = 0; q < 4; ++q) {
        const int col = (cq * 4 + q) * 16 + m;            // channel (B col)
        v2f bb;
        bb.x = book[(size_t)(kb + koff) * C_ + col];
        bb.y = book[(size_t)(kb + koff + 1) * C_ + col];
        acc2[q] = __builtin_amdgcn_wmma_f32_16x16x4_f32(
            false, a, false, bb, (short)0, acc2[q], false, false);
      }
    }
    #pragma unroll
    for (int q = 0; q < 4; ++q) {
      const int col = (cq * 4 + q) * 16 + m;
      #pragma unroll
      for (int v = 0; v < 8; ++v) {
        const int row = rt2 * 16 + v + ((lane >> 4) << 3);
        zt[row * ZLD + col] = acc2[q][v];                 // zt reused as z_q staging
      }
    }
  }
  __syncthreads();

  // write z_q back in [B, C, T] layout, coalesced over t
  #pragma unroll
  for (int it = 0; it < 32; ++it) {
    const int c = it * 8 + (tid >> 5);
    out[ZQ_OFF + ((size_t)(b * C_ + c)) * T_ + t0 + r] = zt[r * ZLD + c];
  }
}

// ---- kernel 2: mean_prob = acc / N ----
__global__ void gvq_finish(const float* __restrict__ ws_macc, float* __restrict__ out) {
  const int k = blockIdx.x * blockDim.x + threadIdx.x;
  if (k < K_) out[MPROB_OFF + k] = ws_macc[k] * (1.0f / (float)N_);
}

extern "C" void kernel_launch(void* const* d_in, const int* in_sizes, int n_in,
                              void* d_out, int out_size, void* d_ws, size_t ws_size,
                              hipStream_t stream) {
  const float* z    = (const float*)d_in[0];
  const float* book = (const float*)d_in[1];
  const float* lpq  = (const float*)d_in[2];
  const float* u    = (const float*)d_in[3];
  // d_in[4] = is_train (train path implemented)
  float* out = (float*)d_out;
  float* ws_bsq  = (float*)d_ws;        // 1024 floats: ||e_k||^2
  float* ws_macc = ws_bsq + K_;         // 1024 floats: mean_prob accumulator

  gvq_prep<<<K_, 256, 0, stream>>>(book, lpq, ws_bsq, ws_macc, out + PQ_OFF);

  const size_t shmem = (size_t)(ROWS * ZLD + ROWS + ROWS * LLD) * sizeof(float);
  gvq_main<<<N_ / ROWS, 256, shmem, stream>>>(z, book, lpq, u, ws_bsq, ws_macc, out);

  gvq_finish<<<K_ / 256, 256, 0, stream>>>(ws_macc, out);
}